// DistanceModule_16501264351949
// MI455X (gfx1250) — compile-verified
//
#include <hip/hip_runtime.h>
#include <math.h>

// Shapes fixed by the reference: B=4, N=512, C=64  -> B*N = 2048 rows.
#define CDIM   64
#define NROWS  2048
#define NSEQ   512
#define LN_EPS 1e-5f

typedef __attribute__((ext_vector_type(2))) float v2f;
typedef __attribute__((ext_vector_type(8))) float v8f;

// ---------------------------------------------------------------------------
// Kernel 1: x = LayerNorm(ReLU(h @ W^T + b))  using V_WMMA_F32_16X16X4_F32.
// Block = 128 threads (4 waves). Block handles 16 rows; wave w handles the
// 16-column output tile [w*16, w*16+16). K=64 walked in 16 steps of 4.
// ---------------------------------------------------------------------------
__global__ __launch_bounds__(128) void mlp_norm_wmma_kernel(
    const float* __restrict__ h, const float* __restrict__ W,
    const float* __restrict__ bias, const float* __restrict__ gamma,
    const float* __restrict__ beta, float* __restrict__ xout)
{
    __shared__ float xs[16 * 65];   // 16 rows x 64 cols, padded stride 65
    __shared__ float mu_s[16];
    __shared__ float rs_s[16];

    const int tid  = threadIdx.x;
    const int wave = tid >> 5;
    const int lane = tid & 31;
    const int m    = lane & 15;     // row index (A) == col index (B) per lane
    const int kh   = lane >> 4;     // K-half selector (0: K=0,1 ; 1: K=2,3)
    const int row0 = blockIdx.x * 16;
    const int n0   = wave * 16;

    const float* __restrict__ arow = h + (row0 + m) * CDIM; // A: h rows
    const float* __restrict__ brow = W + (n0 + m) * CDIM;   // B[k][n] = W[n][k]

    v8f acc = {};
#pragma unroll
    for (int kk = 0; kk < 16; ++kk) {
        const int kb = kk * 4 + 2 * kh;
        v2f a, bf;
        a.x  = arow[kb];     a.y  = arow[kb + 1];
        bf.x = brow[kb];     bf.y = brow[kb + 1];
        // D = A(16x4,f32) * B(4x16,f32) + C  -> v_wmma_f32_16x16x4_f32
        acc = __builtin_amdgcn_wmma_f32_16x16x4_f32(
            false, a, false, bf, (short)0, acc, false, false);
    }

    // Bias + ReLU, scatter D fragment into LDS. Lane holds N = n0+m,
    // VGPR v holds M = v + 8*kh.
    const float bn = bias[n0 + m];
#pragma unroll
    for (int v = 0; v < 8; ++v) {
        const int mr = v + 8 * kh;
        xs[mr * 65 + n0 + m] = fmaxf(acc[v] + bn, 0.0f);
    }
    __syncthreads();

    // Per-row mean / rsqrt(var) over 64 channels (threads 0..15, one row each;
    // padded stride keeps the 16 readers on distinct banks).
    if (tid < 16) {
        float s = 0.0f;
        for (int k = 0; k < CDIM; ++k) s += xs[tid * 65 + k];
        const float mu = s * (1.0f / CDIM);
        float vs = 0.0f;
        for (int k = 0; k < CDIM; ++k) {
            const float d = xs[tid * 65 + k] - mu;
            vs += d * d;
        }
        mu_s[tid] = mu;
        rs_s[tid] = rsqrtf(vs * (1.0f / CDIM) + LN_EPS);
    }
    __syncthreads();

    // Normalize + affine, write x to global scratch. 128 threads x 8 elems.
    const int row = tid >> 3;
    const int cb  = (tid & 7) * 8;
    const float mu = mu_s[row];
    const float rs = rs_s[row];
    float* __restrict__ orow = xout + (row0 + row) * CDIM;
#pragma unroll
    for (int k = 0; k < 8; ++k) {
        const int col = cb + k;
        orow[col] = (xs[row * 65 + col] - mu) * rs * gamma[col] + beta[col];
    }
}

// ---------------------------------------------------------------------------
// Kernel 2: out[b,i,j,:] = softmax_c( x[b,i,:] * x[b,j,:] )
// Pure streaming: 256 MB of stores is the roofline. One wave per (b,i,j),
// float2 per lane (64 channels / 32 lanes), shfl_xor reductions, v_exp_f32.
// grid = (2048, 8), block = 256 (8 waves); each wave covers 8 consecutive j.
// ---------------------------------------------------------------------------
__global__ __launch_bounds__(256) void softmax_outer_kernel(
    const float* __restrict__ x, float* __restrict__ out)
{
    const int bi   = blockIdx.x;          // b*512 + i
    const int lane = threadIdx.x & 31;
    const int wave = threadIdx.x >> 5;
    const int b    = bi >> 9;

    const float2 vi = ((const float2*)(x + (size_t)bi * CDIM))[lane];
    const int jbase = blockIdx.y * 64 + wave * 8;

#pragma unroll
    for (int t = 0; t < 8; ++t) {
        const int j = jbase + t;
        const float2 vj =
            ((const float2*)(x + ((size_t)(b << 9) + j) * CDIM))[lane];

        const float p0 = vi.x * vj.x;
        const float p1 = vi.y * vj.y;

        float m = fmaxf(p0, p1);
#pragma unroll
        for (int off = 16; off > 0; off >>= 1)
            m = fmaxf(m, __shfl_xor(m, off, 32));

        const float e0 = __expf(p0 - m);
        const float e1 = __expf(p1 - m);
        float s = e0 + e1;
#pragma unroll
        for (int off = 16; off > 0; off >>= 1)
            s += __shfl_xor(s, off, 32);

        const float inv = __builtin_amdgcn_rcpf(s);
        float2 o;
        o.x = e0 * inv;
        o.y = e1 * inv;
        ((float2*)(out + (((size_t)bi * NSEQ) + j) * CDIM))[lane] = o;
    }
}

extern "C" void kernel_launch(void* const* d_in, const int* in_sizes, int n_in,
                              void* d_out, int out_size, void* d_ws, size_t ws_size,
                              hipStream_t stream)
{
    (void)in_sizes; (void)n_in; (void)out_size; (void)ws_size;
    const float* h     = (const float*)d_in[0];
    const float* W     = (const float*)d_in[1];
    const float* bias  = (const float*)d_in[2];
    const float* gamma = (const float*)d_in[3];
    const float* beta  = (const float*)d_in[4];

    float* x   = (float*)d_ws;            // 2048*64 f32 = 512 KB scratch
    float* out = (float*)d_out;           // 4*512*512*64 f32 = 256 MB

    mlp_norm_wmma_kernel<<<NROWS / 16, 128, 0, stream>>>(h, W, bias, gamma, beta, x);

    dim3 g2(NROWS, NSEQ / 64);
    softmax_outer_kernel<<<g2, 256, 0, stream>>>(x, out);
}